// MixAttention_7911329759990
// MI455X (gfx1250) — compile-verified
//
#include <hip/hip_runtime.h>
#include <hip/hip_bf16.h>
#include <math.h>

// ---------------- problem constants (match reference) ----------------
#define TT   1024
#define BB   32
#define CC   1024     // Q1=K1=V1=Q2=K2=V2
#define KWW  3
#define NQK  (CC*KWW) // 3072
#define EPSF 1e-6f

typedef float v2f __attribute__((ext_vector_type(2)));
typedef float v8f __attribute__((ext_vector_type(8)));

// ---------------- helpers ----------------
__device__ __forceinline__ float fast_tanhf(float x) {
#if defined(__has_builtin)
#if __has_builtin(__builtin_amdgcn_tanhf)
    return __builtin_amdgcn_tanhf(x);   // v_tanh_f32 (CDNA5 trans op)
#else
    return tanhf(x);
#endif
#else
    return tanhf(x);
#endif
}

__device__ __forceinline__ float wave_sum(float v) {
#pragma unroll
    for (int o = 16; o; o >>= 1) v += __shfl_xor(v, o, 32);
    return v;
}
__device__ __forceinline__ float wave_max(float v) {
#pragma unroll
    for (int o = 16; o; o >>= 1) v = fmaxf(v, __shfl_xor(v, o, 32));
    return v;
}

// ============================================================================
// Kernel 1: small-M GEMM with V_WMMA_F32_16X16X4_F32.
//   out[m,n] = sum_k q[m,k]*W[n,k] + bias[n],  M=32, K=1024, N multiple of 16.
// Per-wave 16x16 output tile; K consumed 4 at a time.
// A layout (16x4 f32):  lanes 0-15: M=lane, K={0,1}; lanes 16-31: M=lane-16, K={2,3}
// B layout (4x16 f32):  lanes 0-15: N=lane, K={0,1}; lanes 16-31: N=lane-16, K={2,3}
// C/D layout: VGPR r -> (M=r, N=lane) for lanes 0-15, (M=r+8, N=lane-16) for 16-31
// ============================================================================
__global__ void wmma_gemm_kernel(const float* __restrict__ q,   // [32, K]
                                 const float* __restrict__ W,   // [N, K]
                                 const float* __restrict__ bias,// [N]
                                 float* __restrict__ out,       // [32, N]
                                 int N, int K)
{
    const int lane  = threadIdx.x & 31;
    const int wave  = threadIdx.x >> 5;
    const int tile  = blockIdx.x * (blockDim.x >> 5) + wave;
    const int nTilesN = N >> 4;
    if (tile >= 2 * nTilesN) return;          // whole-wave uniform exit

    const int mTile = tile / nTilesN;         // 0 or 1 (M=32)
    const int nTile = tile % nTilesN;
    const int m0 = mTile << 4;
    const int n0 = nTile << 4;

    const int half = lane >> 4;               // 0: K pair {0,1}, 1: {2,3}
    const int l    = lane & 15;

    const float* __restrict__ qrow = q + (size_t)(m0 + l) * K + (half << 1);
    const float* __restrict__ wrow = W + (size_t)(n0 + l) * K + (half << 1);

    v8f c = {};
    for (int k0 = 0; k0 < K; k0 += 4) {
        v2f a = *(const v2f*)(qrow + k0);
        v2f b = *(const v2f*)(wrow + k0);
        c = __builtin_amdgcn_wmma_f32_16x16x4_f32(
                /*neg_a=*/false, a, /*neg_b=*/false, b,
                /*c_mod=*/(short)0, c, /*reuse_a=*/false, /*reuse_b=*/false);
    }

    const int n = n0 + l;
    const float bn = bias[n];
#pragma unroll
    for (int r = 0; r < 8; ++r) {
        const int m = m0 + r + (half << 3);
        out[(size_t)m * N + n] = c[r] + bn;
    }
}

// ============================================================================
// Kernel 2: additive-attention logits. One wave per (t,b).
//   a1[t,b] = sum_c tanh(saq[b,c] + k1[t,b,c]) * a1w[c] + a1b
// ============================================================================
__global__ void add_logits_kernel(const float* __restrict__ saq, // [B, C]
                                  const float* __restrict__ k1,  // [T, B, C]
                                  const float* __restrict__ a1w, // [C]
                                  const float* __restrict__ a1b, // [1]
                                  float* __restrict__ a1)        // [T, B]
{
    const int gw = blockIdx.x * (blockDim.x >> 5) + (threadIdx.x >> 5);
    if (gw >= TT * BB) return;
    const int t = gw / BB, b = gw % BB;
    const int lane = threadIdx.x & 31;

    const float* __restrict__ krow = k1 + ((size_t)t * BB + b) * CC;
    const float* __restrict__ srow = saq + (size_t)b * CC;

    float acc = 0.f;
#pragma unroll 4
    for (int c = lane; c < CC; c += 32)
        acc = fmaf(fast_tanhf(srow[c] + krow[c]), a1w[c], acc);

    acc = wave_sum(acc);
    if (lane == 0) a1[(size_t)t * BB + b] = acc + a1b[0];
}

// ============================================================================
// Kernel 3: dynamic grouped conv1d logits. One wave per (t,b).
//   a2[t,b] = sum_c k2[t-1,b,c]*ker[b,c,0] + k2[t,b,c]*ker[b,c,1]
//                 + k2[t+1,b,c]*ker[b,c,2]        (zero-padded at ends)
// ============================================================================
__global__ void conv_logits_kernel(const float* __restrict__ kern, // [B, C, 3]
                                   const float* __restrict__ k2,   // [T, B, C]
                                   float* __restrict__ a2)         // [T, B]
{
    const int gw = blockIdx.x * (blockDim.x >> 5) + (threadIdx.x >> 5);
    if (gw >= TT * BB) return;
    const int t = gw / BB, b = gw % BB;
    const int lane = threadIdx.x & 31;

    const float* __restrict__ kb  = kern + (size_t)b * CC * KWW;
    const float* __restrict__ r0  = k2 + ((size_t)t * BB + b) * CC;
    const float* __restrict__ rm1 = r0 - (size_t)BB * CC;
    const float* __restrict__ rp1 = r0 + (size_t)BB * CC;
    const bool hasM = (t > 0), hasP = (t < TT - 1);

    float acc = 0.f;
#pragma unroll 2
    for (int c = lane; c < CC; c += 32) {
        const float w0 = kb[c * 3 + 0];
        const float w1 = kb[c * 3 + 1];
        const float w2 = kb[c * 3 + 2];
        float s = r0[c] * w1;
        if (hasM) s = fmaf(rm1[c], w0, s);
        if (hasP) s = fmaf(rp1[c], w2, s);
        acc += s;
    }
    acc = wave_sum(acc);
    if (lane == 0) a2[(size_t)t * BB + b] = acc;
}

// ============================================================================
// Kernel 4: masked softmax over T for one batch column. Block = 256 thr per b.
//   e = exp(x - max(x)) * mask;  e /= sum(e)
// ============================================================================
__global__ void softmax_col_kernel(const float* __restrict__ logits, // [T, B]
                                   const float* __restrict__ mask,   // [T, B]
                                   float* __restrict__ e)            // [T, B]
{
    __shared__ float smA[8];
    __shared__ float smB[8];
    const int b = blockIdx.x;
    const int tid = threadIdx.x, lane = tid & 31, w = tid >> 5;

    float x[4];
    float lmax = -3.4e38f;
#pragma unroll
    for (int i = 0; i < 4; ++i) {
        x[i] = logits[(size_t)(tid + 256 * i) * BB + b];
        lmax = fmaxf(lmax, x[i]);
    }
    lmax = wave_max(lmax);
    if (lane == 0) smA[w] = lmax;
    __syncthreads();
    float gmax = smA[0];
#pragma unroll
    for (int i = 1; i < 8; ++i) gmax = fmaxf(gmax, smA[i]);

    float ex[4];
    float lsum = 0.f;
#pragma unroll
    for (int i = 0; i < 4; ++i) {
        ex[i] = __expf(x[i] - gmax) * mask[(size_t)(tid + 256 * i) * BB + b];
        lsum += ex[i];
    }
    lsum = wave_sum(lsum);
    if (lane == 0) smB[w] = lsum;
    __syncthreads();
    float gsum = 0.f;
#pragma unroll
    for (int i = 0; i < 8; ++i) gsum += smB[i];

    const float inv = 1.0f / gsum;
#pragma unroll
    for (int i = 0; i < 4; ++i)
        e[(size_t)(tid + 256 * i) * BB + b] = ex[i] * inv;
}

// ============================================================================
// Kernel 5: partial weighted sums of v1/v2 over a T-chunk of 256.
//   partial[ts][b][d] = sum_{t in chunk} e1[t,b]*v1[t,b,d] + e2[t,b]*v2[t,b,d]
// grid = (V/128, B, 4), block = 128. e-columns staged in LDS; v loads coalesced.
// ============================================================================
__global__ void attend_partial_kernel(const float* __restrict__ e1,
                                      const float* __restrict__ v1,
                                      const float* __restrict__ e2,
                                      const float* __restrict__ v2,
                                      float* __restrict__ partial) // [4, B, C]
{
    __shared__ float se1[256];
    __shared__ float se2[256];
    const int b  = blockIdx.y;
    const int ts = blockIdx.z;
    const int t0 = ts * 256;

    for (int i = threadIdx.x; i < 256; i += blockDim.x) {
        se1[i] = e1[(size_t)(t0 + i) * BB + b];
        se2[i] = e2[(size_t)(t0 + i) * BB + b];
    }
    __syncthreads();

    const int d = blockIdx.x * blockDim.x + threadIdx.x;
    const size_t stride = (size_t)BB * CC;
    const float* __restrict__ p1 = v1 + ((size_t)t0 * BB + b) * CC + d;
    const float* __restrict__ p2 = v2 + ((size_t)t0 * BB + b) * CC + d;

    float acc = 0.f;
#pragma unroll 4
    for (int i = 0; i < 256; ++i) {
        acc = fmaf(se1[i], p1[(size_t)i * stride], acc);
        acc = fmaf(se2[i], p2[(size_t)i * stride], acc);
    }
    partial[((size_t)ts * BB + b) * CC + d] = acc;
}

// ============================================================================
// Kernel 6: sum partials + LayerNorm over feature dim. Block = 256 thr per b.
// ============================================================================
__global__ void finalize_ln_kernel(const float* __restrict__ partial, // [4,B,C]
                                   const float* __restrict__ g,       // [C]
                                   const float* __restrict__ beta,    // [C]
                                   float* __restrict__ out)           // [B, C]
{
    __shared__ float smS[8];
    __shared__ float smQ[8];
    const int b = blockIdx.x;
    const int tid = threadIdx.x, lane = tid & 31, w = tid >> 5;
    const size_t cb = (size_t)BB * CC;

    float xv[4];
    float s = 0.f, s2 = 0.f;
#pragma unroll
    for (int i = 0; i < 4; ++i) {
        const int d = tid + 256 * i;
        float x = partial[(size_t)b * CC + d]
                + partial[cb     + (size_t)b * CC + d]
                + partial[2 * cb + (size_t)b * CC + d]
                + partial[3 * cb + (size_t)b * CC + d];
        xv[i] = x;
        s += x;
        s2 = fmaf(x, x, s2);
    }
    s  = wave_sum(s);
    s2 = wave_sum(s2);
    if (lane == 0) { smS[w] = s; smQ[w] = s2; }
    __syncthreads();
    float gs = 0.f, gq = 0.f;
#pragma unroll
    for (int i = 0; i < 8; ++i) { gs += smS[i]; gq += smQ[i]; }

    const float mu  = gs * (1.0f / CC);
    const float var = gq * (1.0f / CC) - mu * mu;
    const float inv = rsqrtf(var + EPSF);
#pragma unroll
    for (int i = 0; i < 4; ++i) {
        const int d = tid + 256 * i;
        out[(size_t)b * CC + d] = (xv[i] - mu) * inv * g[d] + beta[d];
    }
}

// ============================================================================
// launch
// ============================================================================
extern "C" void kernel_launch(void* const* d_in, const int* in_sizes, int n_in,
                              void* d_out, int out_size, void* d_ws, size_t ws_size,
                              hipStream_t stream)
{
    const float* q1    = (const float*)d_in[0];   // [B, C]
    const float* k1    = (const float*)d_in[1];   // [T, B, C]
    const float* v1    = (const float*)d_in[2];   // [T, B, C]
    const float* q2    = (const float*)d_in[3];   // [B, C]
    const float* k2    = (const float*)d_in[4];   // [T, B, C]
    const float* v2    = (const float*)d_in[5];   // [T, B, C]
    const float* kmask = (const float*)d_in[6];   // [T, B]
    const float* sa_w  = (const float*)d_in[7];   // [C, C]
    const float* sa_b  = (const float*)d_in[8];   // [C]
    const float* a1_w  = (const float*)d_in[9];   // [1, C]
    const float* a1_b  = (const float*)d_in[10];  // [1]
    const float* qk_w  = (const float*)d_in[11];  // [C*3, C]
    const float* qk_b  = (const float*)d_in[12];  // [C*3]
    const float* ln_g  = (const float*)d_in[13];  // [C]
    const float* ln_b  = (const float*)d_in[14];  // [C]
    float* out = (float*)d_out;                   // [B, C]

    // ---- workspace layout (floats) ----
    float* ws      = (float*)d_ws;
    float* saq     = ws;                          // 32*1024      = 32768
    float* kern    = saq  + BB * CC;              // 32*3072      = 98304
    float* a1log   = kern + BB * NQK;             // 32768
    float* a2log   = a1log + TT * BB;             // 32768
    float* e1      = a2log + TT * BB;             // 32768
    float* e2      = e1    + TT * BB;             // 32768
    float* partial = e2    + TT * BB;             // 4*32*1024    = 131072

    // 1) projections via V_WMMA_F32_16X16X4_F32
    //    saq  = q1 @ sa_w.T + sa_b  -> tiles = 2*(1024/16) = 128 waves = 32 blocks
    wmma_gemm_kernel<<<32, 128, 0, stream>>>(q1, sa_w, sa_b, saq, CC, CC);
    //    kern = q2 @ qk_w.T + qk_b  -> tiles = 2*(3072/16) = 384 waves = 96 blocks
    wmma_gemm_kernel<<<96, 128, 0, stream>>>(q2, qk_w, qk_b, kern, NQK, CC);

    // 2) additive logits: one wave per (t,b) -> 32768 waves, 8 per block
    add_logits_kernel<<<(TT * BB) / 8, 256, 0, stream>>>(saq, k1, a1_w, a1_b, a1log);

    // 3) dynamic-conv logits
    conv_logits_kernel<<<(TT * BB) / 8, 256, 0, stream>>>(kern, k2, a2log);

    // 4) masked softmax over T (per batch column)
    softmax_col_kernel<<<BB, 256, 0, stream>>>(a1log, kmask, e1);
    softmax_col_kernel<<<BB, 256, 0, stream>>>(a2log, kmask, e2);

    // 5) fused weighted sums of v1 & v2, T split into 4 chunks
    attend_partial_kernel<<<dim3(CC / 128, BB, 4), 128, 0, stream>>>(e1, v1, e2, v2, partial);

    // 6) reduce partials + LayerNorm
    finalize_ln_kernel<<<BB, 256, 0, stream>>>(partial, ln_g, ln_b, out);
}